// GATModel_56942676410568
// MI455X (gfx1250) — compile-verified
//
#include <hip/hip_runtime.h>
#include <hip/hip_bf16.h>

// ---------------------------------------------------------------------------
// GAT (2-layer, DGL-style) for MI455X / gfx1250.
// fp32 throughout (GEMMs are not the bottleneck; edge gather/scatter through
// the 192MB L2 is). GEMMs use V_WMMA_F32_16X16X4_F32.
// ---------------------------------------------------------------------------

#define N_NODES 50000
#define N_EDGES 800000
#define IN_FEATS 128
#define HIDDEN 64
#define HEADS 4
#define FEAT1 (HEADS * HIDDEN)   // 256
#define NUM_CLASSES 16
#define NEG_SLOPE 0.2f

typedef float v2f __attribute__((ext_vector_type(2)));
typedef float v8f __attribute__((ext_vector_type(8)));

// ---- order-preserving float <-> uint mapping for atomicMax segment-max ----
__device__ __forceinline__ unsigned fenc(float f) {
  unsigned u = __float_as_uint(f);
  return (u & 0x80000000u) ? ~u : (u | 0x80000000u);
}
__device__ __forceinline__ float fdec(unsigned k) {
  unsigned u = (k & 0x80000000u) ? (k & 0x7FFFFFFFu) : ~k;
  return __uint_as_float(u);
}
__device__ __forceinline__ float leaky(float v) { return v > 0.0f ? v : NEG_SLOPE * v; }

// ---------------------------------------------------------------------------
// fills
// ---------------------------------------------------------------------------
__global__ void fill_f(float* __restrict__ p, float v, int n) {
  int i = blockIdx.x * blockDim.x + threadIdx.x;
  if (i < n) p[i] = v;
}
__global__ void fill_u(unsigned* __restrict__ p, unsigned v, int n) {
  int i = blockIdx.x * blockDim.x + threadIdx.x;
  if (i < n) p[i] = v;
}
__global__ void init_out(float* __restrict__ out, const float* __restrict__ b2, int n) {
  int i = blockIdx.x * blockDim.x + threadIdx.x;
  if (i < n) out[i] = b2[i & (NUM_CLASSES - 1)];
}

// ---------------------------------------------------------------------------
// GEMM1: H1[N,256] = X[N,128] @ W1[128,256], fp32 WMMA 16x16x4.
// block = 256 threads = 8 waves; wave -> (mtile = by*2 + w/4, ntile = bx*4 + w%4)
// grid = (4, ceil(3125/2)=1563)
// ---------------------------------------------------------------------------
__global__ __launch_bounds__(256) void gemm1_wmma(const float* __restrict__ X,
                                                  const float* __restrict__ W,
                                                  float* __restrict__ H1) {
  const int lane = threadIdx.x & 31;
  const int wave = threadIdx.x >> 5;
  const int m0 = (blockIdx.y * 2 + (wave >> 2)) * 16;
  const int n0 = (blockIdx.x * 4 + (wave & 3)) * 16;
  const int half = lane >> 4;   // selects K pair {0,1} vs {2,3}
  const int l = lane & 15;

  int arow = m0 + l;
  if (arow > N_NODES - 1) arow = N_NODES - 1;   // clamp: keep EXEC all-1s

  v8f acc = {};
  for (int k0 = 0; k0 < IN_FEATS; k0 += 4) {
    const int ka = k0 + 2 * half;
    v2f a, b;
    a.x = X[arow * IN_FEATS + ka];
    a.y = X[arow * IN_FEATS + ka + 1];
    b.x = W[ka * FEAT1 + n0 + l];
    b.y = W[(ka + 1) * FEAT1 + n0 + l];
    acc = __builtin_amdgcn_wmma_f32_16x16x4_f32(false, a, false, b, (short)0, acc,
                                                false, false);
  }
  const int col = n0 + l;
  const int rbase = m0 + half * 8;
#pragma unroll
  for (int j = 0; j < 8; ++j) {
    const int r = rbase + j;
    if (r < N_NODES) H1[r * FEAT1 + col] = acc[j];
  }
}

// ---------------------------------------------------------------------------
// GEMM2: G2[N,16] = H2[N,256] @ W2[256,16], single col tile, wave per row tile.
// block = 256 = 8 waves; grid = ceil(3125/8) = 391
// ---------------------------------------------------------------------------
__global__ __launch_bounds__(256) void gemm2_wmma(const float* __restrict__ H2,
                                                  const float* __restrict__ W2,
                                                  float* __restrict__ G2) {
  const int lane = threadIdx.x & 31;
  const int wave = threadIdx.x >> 5;
  const int m0 = (blockIdx.x * 8 + wave) * 16;
  const int half = lane >> 4;
  const int l = lane & 15;

  int arow = m0 + l;
  if (arow > N_NODES - 1) arow = N_NODES - 1;

  v8f acc = {};
  for (int k0 = 0; k0 < FEAT1; k0 += 4) {
    const int ka = k0 + 2 * half;
    v2f a, b;
    a.x = H2[arow * FEAT1 + ka];
    a.y = H2[arow * FEAT1 + ka + 1];
    b.x = W2[ka * NUM_CLASSES + l];
    b.y = W2[(ka + 1) * NUM_CLASSES + l];
    acc = __builtin_amdgcn_wmma_f32_16x16x4_f32(false, a, false, b, (short)0, acc,
                                                false, false);
  }
  const int rbase = m0 + half * 8;
#pragma unroll
  for (int j = 0; j < 8; ++j) {
    const int r = rbase + j;
    if (r < N_NODES) G2[r * NUM_CLASSES + l] = acc[j];
  }
}

// ---------------------------------------------------------------------------
// layer-1 attention scalars: el/er[n*4+h] = <h1[n,h,:], attn[h,:]>
// ---------------------------------------------------------------------------
__global__ void attn_dot1(const float* __restrict__ H1, const float* __restrict__ al,
                          const float* __restrict__ ar, float* __restrict__ el,
                          float* __restrict__ er) {
  int t = blockIdx.x * blockDim.x + threadIdx.x;
  if (t >= N_NODES * HEADS) return;
  const int h = t & (HEADS - 1);
  const int n = t >> 2;
  const float* hv = H1 + n * FEAT1 + h * HIDDEN;
  const float* av = al + h * HIDDEN;
  const float* bv = ar + h * HIDDEN;
  float sl = 0.f, sr = 0.f;
#pragma unroll 8
  for (int d = 0; d < HIDDEN; ++d) {
    const float x = hv[d];
    sl += x * av[d];
    sr += x * bv[d];
  }
  el[t] = sl;
  er[t] = sr;
}

// ---- layer-1 edge passes (thread per edge*head) ----
__global__ void edge_max1(const float* __restrict__ el, const float* __restrict__ er,
                          const int* __restrict__ src, const int* __restrict__ dst,
                          float* __restrict__ ebuf, unsigned* __restrict__ m) {
  int t = blockIdx.x * blockDim.x + threadIdx.x;
  if (t >= N_EDGES * HEADS) return;
  const int h = t & (HEADS - 1);
  const int e = t >> 2;
  const int s = src[e], d = dst[e];
  const float v = leaky(el[s * HEADS + h] + er[d * HEADS + h]);
  ebuf[t] = v;
  atomicMax(&m[d * HEADS + h], fenc(v));
}

__global__ void edge_exp1(const int* __restrict__ dst, float* __restrict__ ebuf,
                          const unsigned* __restrict__ m, float* __restrict__ ssum) {
  int t = blockIdx.x * blockDim.x + threadIdx.x;
  if (t >= N_EDGES * HEADS) return;
  const int h = t & (HEADS - 1);
  const int e = t >> 2;
  const int d = dst[e];
  const float ex = __expf(ebuf[t] - fdec(m[d * HEADS + h]));
  ebuf[t] = ex;
  atomicAdd(&ssum[d * HEADS + h], ex);
}

__global__ void edge_norm1(const int* __restrict__ dst, float* __restrict__ ebuf,
                           const float* __restrict__ ssum) {
  int t = blockIdx.x * blockDim.x + threadIdx.x;
  if (t >= N_EDGES * HEADS) return;
  const int h = t & (HEADS - 1);
  const int e = t >> 2;
  ebuf[t] = ebuf[t] / ssum[dst[e] * HEADS + h];
}

// wave-per-edge scatter-add of alpha * h1[src] into agg[dst]  (256 feats)
__global__ __launch_bounds__(256) void aggregate1(const float* __restrict__ H1,
                                                  const float* __restrict__ alpha,
                                                  const int* __restrict__ src,
                                                  const int* __restrict__ dst,
                                                  float* __restrict__ agg) {
  const int e = blockIdx.x * 8 + (threadIdx.x >> 5);
  if (e >= N_EDGES) return;
  const int lane = threadIdx.x & 31;
  const int s = src[e], d = dst[e];
#pragma unroll
  for (int st = 0; st < 8; ++st) {
    const int c = st * 32 + lane;
    const float a = alpha[e * HEADS + (c >> 6)];   // wave-uniform per step
    atomicAdd(&agg[d * FEAT1 + c], a * H1[s * FEAT1 + c]);
  }
}

// elu(agg + b1) in place -> layer-2 input
__global__ void finalize1(float* __restrict__ agg, const float* __restrict__ b1) {
  int t = blockIdx.x * blockDim.x + threadIdx.x;
  if (t >= N_NODES * FEAT1) return;
  const float x = agg[t] + b1[t & (FEAT1 - 1)];
  agg[t] = x > 0.f ? x : (__expf(x) - 1.f);
}

// ---------------------------------------------------------------------------
// layer-2 (single head, 16 classes)
// ---------------------------------------------------------------------------
__global__ void attn_dot2(const float* __restrict__ G2, const float* __restrict__ al,
                          const float* __restrict__ ar, float* __restrict__ el,
                          float* __restrict__ er) {
  int n = blockIdx.x * blockDim.x + threadIdx.x;
  if (n >= N_NODES) return;
  const float* g = G2 + n * NUM_CLASSES;
  float sl = 0.f, sr = 0.f;
#pragma unroll
  for (int c = 0; c < NUM_CLASSES; ++c) {
    sl += g[c] * al[c];
    sr += g[c] * ar[c];
  }
  el[n] = sl;
  er[n] = sr;
}

__global__ void edge_max2(const float* __restrict__ el, const float* __restrict__ er,
                          const int* __restrict__ src, const int* __restrict__ dst,
                          float* __restrict__ ebuf, unsigned* __restrict__ m) {
  int e = blockIdx.x * blockDim.x + threadIdx.x;
  if (e >= N_EDGES) return;
  const float v = leaky(el[src[e]] + er[dst[e]]);
  ebuf[e] = v;
  atomicMax(&m[dst[e]], fenc(v));
}

__global__ void edge_exp2(const int* __restrict__ dst, float* __restrict__ ebuf,
                          const unsigned* __restrict__ m, float* __restrict__ ssum) {
  int e = blockIdx.x * blockDim.x + threadIdx.x;
  if (e >= N_EDGES) return;
  const float ex = __expf(ebuf[e] - fdec(m[dst[e]]));
  ebuf[e] = ex;
  atomicAdd(&ssum[dst[e]], ex);
}

__global__ void edge_norm2(const int* __restrict__ dst, float* __restrict__ ebuf,
                           const float* __restrict__ ssum) {
  int e = blockIdx.x * blockDim.x + threadIdx.x;
  if (e >= N_EDGES) return;
  ebuf[e] = ebuf[e] / ssum[dst[e]];
}

// thread per (edge, class): out[dst, c] += alpha[e] * g2[src, c]
__global__ void aggregate2(const float* __restrict__ G2, const float* __restrict__ alpha,
                           const int* __restrict__ src, const int* __restrict__ dst,
                           float* __restrict__ out) {
  int t = blockIdx.x * blockDim.x + threadIdx.x;
  if (t >= N_EDGES * NUM_CLASSES) return;
  const int c = t & (NUM_CLASSES - 1);
  const int e = t >> 4;
  atomicAdd(&out[dst[e] * NUM_CLASSES + c], alpha[e] * G2[src[e] * NUM_CLASSES + c]);
}

// ---------------------------------------------------------------------------
// launch
// ---------------------------------------------------------------------------
extern "C" void kernel_launch(void* const* d_in, const int* in_sizes, int n_in,
                              void* d_out, int out_size, void* d_ws, size_t ws_size,
                              hipStream_t stream) {
  const float* X  = (const float*)d_in[0];
  const float* W1 = (const float*)d_in[1];
  const float* al1 = (const float*)d_in[2];
  const float* ar1 = (const float*)d_in[3];
  const float* b1  = (const float*)d_in[4];
  const float* W2  = (const float*)d_in[5];
  const float* al2 = (const float*)d_in[6];
  const float* ar2 = (const float*)d_in[7];
  const float* b2  = (const float*)d_in[8];
  const int* src = (const int*)d_in[9];
  const int* dst = (const int*)d_in[10];
  float* out = (float*)d_out;

  float* ws = (float*)d_ws;
  // layer-1 workspace layout (floats)
  float*    H1  = ws;                        // 12,800,000
  float*    AGG = ws + 12800000;             // 12,800,000 (becomes h2 after ELU)
  float*    E1  = ws + 25600000;             //  3,200,000
  float*    EL1 = ws + 28800000;             //    200,000
  float*    ER1 = ws + 29000000;             //    200,000
  unsigned* M1  = (unsigned*)(ws + 29200000);//    200,000
  float*    S1  = ws + 29400000;             //    200,000
  // layer-2 workspace reuses dead H1 region
  float*    G2  = ws;                        //    800,000
  float*    EL2 = ws + 800000;               //     50,000
  float*    ER2 = ws + 850000;               //     50,000
  unsigned* M2  = (unsigned*)(ws + 900000);  //     50,000
  float*    S2  = ws + 950000;               //     50,000
  float*    E2  = ws + 1000000;              //    800,000

  const int T = 256;

  // ---- layer 1 ----
  gemm1_wmma<<<dim3(4, 1563), T, 0, stream>>>(X, W1, H1);

  fill_f<<<(N_NODES * FEAT1 + T - 1) / T, T, 0, stream>>>(AGG, 0.f, N_NODES * FEAT1);
  fill_u<<<(N_NODES * HEADS + T - 1) / T, T, 0, stream>>>(M1, 0u, N_NODES * HEADS);
  fill_f<<<(N_NODES * HEADS + T - 1) / T, T, 0, stream>>>(S1, 0.f, N_NODES * HEADS);

  attn_dot1<<<(N_NODES * HEADS + T - 1) / T, T, 0, stream>>>(H1, al1, ar1, EL1, ER1);
  edge_max1<<<(N_EDGES * HEADS + T - 1) / T, T, 0, stream>>>(EL1, ER1, src, dst, E1, M1);
  edge_exp1<<<(N_EDGES * HEADS + T - 1) / T, T, 0, stream>>>(dst, E1, M1, S1);
  edge_norm1<<<(N_EDGES * HEADS + T - 1) / T, T, 0, stream>>>(dst, E1, S1);
  aggregate1<<<N_EDGES / 8, T, 0, stream>>>(H1, E1, src, dst, AGG);
  finalize1<<<(N_NODES * FEAT1 + T - 1) / T, T, 0, stream>>>(AGG, b1);

  // ---- layer 2 ----
  gemm2_wmma<<<391, T, 0, stream>>>(AGG, W2, G2);

  fill_u<<<(N_NODES + T - 1) / T, T, 0, stream>>>(M2, 0u, N_NODES);
  fill_f<<<(N_NODES + T - 1) / T, T, 0, stream>>>(S2, 0.f, N_NODES);
  init_out<<<(N_NODES * NUM_CLASSES + T - 1) / T, T, 0, stream>>>(out, b2,
                                                                  N_NODES * NUM_CLASSES);

  attn_dot2<<<(N_NODES + T - 1) / T, T, 0, stream>>>(G2, al2, ar2, EL2, ER2);
  edge_max2<<<(N_EDGES + T - 1) / T, T, 0, stream>>>(EL2, ER2, src, dst, E2, M2);
  edge_exp2<<<(N_EDGES + T - 1) / T, T, 0, stream>>>(dst, E2, M2, S2);
  edge_norm2<<<(N_EDGES + T - 1) / T, T, 0, stream>>>(dst, E2, S2);
  aggregate2<<<(N_EDGES * NUM_CLASSES + T - 1) / T, T, 0, stream>>>(G2, E2, src, dst, out);
}